// ScaledDotProductAttention_rmatt_ev_cnn_64493228917088
// MI455X (gfx1250) — compile-verified
//
#include <hip/hip_runtime.h>

// MI455X (gfx1250) fused attention + conv-mix pipeline.
// Roofline: attn (N*L*L f32 = 256MB) is a required OUTPUT, so the problem is
// HBM-bound (~600MB min traffic -> ~26us floor @ 23.3 TB/s) while total FLOPs
// are only ~36G. Exact-precision V_WMMA_F32_16X16X4_F32 drives all three
// GEMMs; the only materialized intermediate is the masked q_rm*k^T score
// tensor (needed by the cross-head 3x3 conv with row halos; halo re-reads are
// L2-served: 32MB/batch working set vs 192MB L2). The attn_pre stream (256MB,
// the largest input) is staged into LDS with gfx1250 async-to-LDS copies
// overlapped with the S-GEMM WMMA phase.
//
// d_ws requirement: N*L*L*4 = 256MB for the masked score intermediate.

#define NB 8
#define NH 8
#define NN 64
#define LL 1024
#define DK 64
#define DV 64
#define ROWS 16
#define ASTR 1032  // padded LDS row stride (floats) to dodge bank conflicts

typedef float v2f __attribute__((ext_vector_type(2)));
typedef float v8f __attribute__((ext_vector_type(8)));

__device__ __forceinline__ float warp_max(float x) {
#pragma unroll
  for (int off = 16; off > 0; off >>= 1) x = fmaxf(x, __shfl_xor(x, off, 32));
  return x;
}
__device__ __forceinline__ float warp_sum(float x) {
#pragma unroll
  for (int off = 16; off > 0; off >>= 1) x += __shfl_xor(x, off, 32);
  return x;
}

// K1: Rmask[n] = where(j>i, 0, (q_rm . k^T) / 8)   -- f32 WMMA, one 16-row
// block per workgroup, 8 waves each own 8 column tiles.
__global__ __launch_bounds__(256) void k_qk_mask(const float* __restrict__ qrm,
                                                 const float* __restrict__ km,
                                                 float* __restrict__ rmask) {
  const int wg = blockIdx.x;
  const int n = wg >> 6;
  const int row0 = (wg & 63) * ROWS;
  const int lane = threadIdx.x & 31;
  const int wave = threadIdx.x >> 5;
  const int m15 = lane & 15;
  const int khalf = (lane >> 4) * 2;  // wmma f32 16x16x4: lanes 16-31 hold K=2,3

  const float* qn = qrm + ((size_t)n * LL + row0 + m15) * DK + khalf;
  const float* kn = km + (size_t)n * LL * DK;
  float* rn = rmask + (size_t)n * LL * LL;

  v2f afrag[16];  // A fragments for all K=64 (16 steps of K=4), reused per tile
#pragma unroll
  for (int s = 0; s < 16; ++s) afrag[s] = *(const v2f*)(qn + 4 * s);

  for (int t = wave; t < 64; t += 8) {
    const int col0 = t * 16;
    const float* kr = kn + (size_t)(col0 + m15) * DK + khalf;
    v8f c = {};
#pragma unroll
    for (int s = 0; s < 16; ++s) {
      v2f b = *(const v2f*)(kr + 4 * s);  // B[k][j] = km[j][k]
      c = __builtin_amdgcn_wmma_f32_16x16x4_f32(false, afrag[s], false, b,
                                                (short)0, c, false, false);
    }
    const int rhi = (lane >> 4) << 3;
#pragma unroll
    for (int g = 0; g < 8; ++g) {
      const int r = row0 + g + rhi;
      const int col = col0 + m15;
      rn[(size_t)r * LL + col] = (col > r) ? 0.0f : c[g] * 0.125f;
    }
  }
}

// K2: per (head, 16-row block): async-stage attn_pre panel to LDS while the
// S GEMM runs on WMMA; coalesced conv/mix/mask pass; softmax; then waves 0-3
// run the attn.V WMMA while waves 4-7 stream attn to global.
__global__ __launch_bounds__(256) void k_attn(
    const float* __restrict__ q, const float* __restrict__ km,
    const float* __restrict__ v, const float* __restrict__ pre,
    const float* __restrict__ cw, const int* __restrict__ fgp,
    const float* __restrict__ rmask, float* __restrict__ outp,
    float* __restrict__ attnp) {
  __shared__ float Arow[ROWS * ASTR];  // 16 x 1024 pre/post-softmax panel
  __shared__ float preS[ROWS * LL];    // attn_pre panel (async-filled)
  __shared__ float wsh[NH * 9];        // conv weights for this output head

  const int wg = blockIdx.x;
  const int n = wg >> 6;
  const int row0 = (wg & 63) * ROWS;
  const int bidx = n >> 3;
  const int h = n & 7;
  const int tid = threadIdx.x;
  const int lane = tid & 31;
  const int wave = tid >> 5;
  const int m15 = lane & 15;
  const int khalf = (lane >> 4) * 2;

  if (tid < NH * 9) wsh[tid] = cw[h * (NH * 9) + tid];  // conv_w[h][c][a][b]

  // ---- Kick off async HBM->LDS copy of the attn_pre row panel (64KB).
  // ASYNCcnt-tracked; overlaps with the Phase A WMMA GEMM below.
  {
    const float* pre_base = pre + ((size_t)n * LL + row0) * LL;
    for (int i = tid; i < ROWS * LL / 4; i += 256) {
      unsigned ldsa = (unsigned)(uintptr_t)(&preS[i * 4]);
      const float* g = pre_base + i * 4;
      asm volatile("global_load_async_to_lds_b128 %0, %1, off"
                   :
                   : "v"(ldsa), "v"(g)
                   : "memory");
    }
  }

  // ---- Phase A: Arow = 0.54 * (q.k^T / 8) via f32 WMMA ----
  {
    const float* qn = q + ((size_t)n * LL + row0 + m15) * DK + khalf;
    const float* kn = km + (size_t)n * LL * DK;
    v2f afrag[16];
#pragma unroll
    for (int s = 0; s < 16; ++s) afrag[s] = *(const v2f*)(qn + 4 * s);
    for (int t = wave; t < 64; t += 8) {
      const int col0 = t * 16;
      const float* kr = kn + (size_t)(col0 + m15) * DK + khalf;
      v8f c = {};
#pragma unroll
      for (int s = 0; s < 16; ++s) {
        v2f b = *(const v2f*)(kr + 4 * s);
        c = __builtin_amdgcn_wmma_f32_16x16x4_f32(false, afrag[s], false, b,
                                                  (short)0, c, false, false);
      }
      const int rhi = (lane >> 4) << 3;
#pragma unroll
      for (int g = 0; g < 8; ++g)
        Arow[(g + rhi) * ASTR + col0 + m15] = 0.0675f * c[g];  // 0.54/TEMP
    }
  }
  asm volatile("s_wait_asynccnt 0x0" ::: "memory");  // preS resident
  __syncthreads();

  // ---- Phase B: += 0.1*pre + 0.324*Rmask + 0.036*T; causal mask ----
  // 4 consecutive columns per thread, lanes contiguous within a row: all
  // attn_pre (LDS), Rmask and conv-tap accesses are coalesced. T =
  // relu(conv3x3(Rmask, 8ch)) shifted down(-right); boundaries fall back to
  // Rmask. Fully-masked 4-col groups short-circuit to -inf (causal skip).
  {
    const int fg = *fgp;
    const int joff = (fg == 1) ? 1 : 0;  // sift_down_right vs sift_down
    const int j0 = tid * 4;
    const float* rm_n = rmask + (size_t)n * LL * LL;
    const float* convbase = rmask + (size_t)(bidx * NH) * LL * LL;
    for (int rl = 0; rl < ROWS; ++rl) {
      const int rg = row0 + rl;
      float* arow = &Arow[rl * ASTR];
      if (j0 > rg) {
#pragma unroll
        for (int jj = 0; jj < 4; ++jj) arow[j0 + jj] = -__builtin_inff();
        continue;
      }
      float acc[4] = {0.0f, 0.0f, 0.0f, 0.0f};
      if (rg >= 1) {
        for (int ch = 0; ch < NH; ++ch) {
          const float* inch = convbase + (size_t)ch * LL * LL;
#pragma unroll
          for (int a = 0; a < 3; ++a) {
            const int ry = rg - 2 + a;  // conv tap rows for x row rg-1
            if (ry < 0) continue;       // SAME zero padding (ry<=rg<=1023)
            const float* inr = inch + (size_t)ry * LL;
            float buf[6];  // tap cols j0-joff-1 .. j0-joff+4
#pragma unroll
            for (int t2 = 0; t2 < 6; ++t2) {
              const int cc = j0 - joff - 1 + t2;
              buf[t2] = (cc >= 0 && cc < LL) ? inr[cc] : 0.0f;
            }
            const float w0 = wsh[ch * 9 + a * 3 + 0];
            const float w1 = wsh[ch * 9 + a * 3 + 1];
            const float w2 = wsh[ch * 9 + a * 3 + 2];
#pragma unroll
            for (int jj = 0; jj < 4; ++jj)
              acc[jj] += w0 * buf[jj] + w1 * buf[jj + 1] + w2 * buf[jj + 2];
          }
        }
      }
      const float* rmr = rm_n + (size_t)rg * LL;
#pragma unroll
      for (int jj = 0; jj < 4; ++jj) {
        const int j = j0 + jj;
        const float Rm = rmr[j];
        const float T =
            (rg == 0 || (joff == 1 && j == 0)) ? Rm : fmaxf(acc[jj], 0.0f);
        const float val =
            arow[j] + 0.1f * preS[rl * LL + j] + 0.324f * Rm + 0.036f * T;
        arow[j] = (j > rg) ? -__builtin_inff() : val;
      }
    }
  }
  __syncthreads();

  // ---- Phase C: row softmax (2 rows per wave, shuffle reductions) ----
#pragma unroll
  for (int rr = 0; rr < 2; ++rr) {
    const int rl = wave * 2 + rr;
    float m = -__builtin_inff();
    for (int j = lane; j < LL; j += 32) m = fmaxf(m, Arow[rl * ASTR + j]);
    m = warp_max(m);
    float s = 0.0f;
    for (int j = lane; j < LL; j += 32) {
      const float e = __expf(Arow[rl * ASTR + j] - m);
      Arow[rl * ASTR + j] = e;
      s += e;
    }
    s = warp_sum(s);
    const float inv = 1.0f / s;
    for (int j = lane; j < LL; j += 32) Arow[rl * ASTR + j] *= inv;
  }
  __syncthreads();

  // ---- Phase D: waves 0-3: out = attn.v (K=1024 WMMA from LDS);
  //               waves 4-7: stream attn panel to global (coalesced) ----
  if (wave < 4) {
    const int d0 = wave * 16;
    const float* vn = v + (size_t)n * LL * DV;
    v8f c = {};
    for (int kk = 0; kk < LL; kk += 4) {
      v2f a = *(const v2f*)(&Arow[m15 * ASTR + kk + khalf]);
      v2f b;
      b.x = vn[(size_t)(kk + khalf) * DV + d0 + m15];
      b.y = vn[(size_t)(kk + khalf + 1) * DV + d0 + m15];
      c = __builtin_amdgcn_wmma_f32_16x16x4_f32(false, a, false, b, (short)0, c,
                                                false, false);
    }
    const int rhi = (lane >> 4) << 3;
#pragma unroll
    for (int g = 0; g < 8; ++g)
      outp[((size_t)n * LL + row0 + g + rhi) * DV + d0 + m15] = c[g];
  } else {
    float* an = attnp + ((size_t)n * LL + row0) * LL;
    for (int idx = tid - 128; idx < ROWS * LL; idx += 128) {
      const int r = idx >> 10;
      const int j = idx & (LL - 1);
      an[(size_t)r * LL + j] = Arow[r * ASTR + j];
    }
  }
}

extern "C" void kernel_launch(void* const* d_in, const int* in_sizes, int n_in,
                              void* d_out, int out_size, void* d_ws,
                              size_t ws_size, hipStream_t stream) {
  (void)in_sizes; (void)n_in; (void)out_size; (void)ws_size;
  const float* q = (const float*)d_in[0];
  const float* k = (const float*)d_in[1];
  const float* v = (const float*)d_in[2];
  const float* qrm = (const float*)d_in[3];
  const float* pre = (const float*)d_in[4];
  const float* cw = (const float*)d_in[5];
  // d_in[6] = mask (recomputed analytically on device), d_in[7] = fg scalar
  const int* fg = (const int*)d_in[7];

  float* outp = (float*)d_out;                 // (N, L, DV)
  float* attnp = outp + (size_t)NN * LL * DV;  // (N, L, L)
  float* rmask = (float*)d_ws;                 // needs 256MB scratch

  const dim3 grid(NN * (LL / ROWS));  // 4096 workgroups
  const dim3 block(256);              // 8 wave32s
  hipLaunchKernelGGL(k_qk_mask, grid, block, 0, stream, qrm, k, rmask);
  hipLaunchKernelGGL(k_attn, grid, block, 0, stream, q, k, v, pre, cw, fg,
                     rmask, outp, attnp);
}